// ChebNet_75694503625267
// MI455X (gfx1250) — compile-verified
//
#include <hip/hip_runtime.h>

typedef float v2f __attribute__((ext_vector_type(2)));
typedef float v8f __attribute__((ext_vector_type(8)));

#define F_IN  500
#define HID   16
#define NCLS  7
#define KSTEP (F_IN / 4)          // 125 WMMA K-steps

// ---------------- degree / normalization ----------------
__global__ void deg_kernel(const int* __restrict__ row, unsigned* __restrict__ deg, int E) {
    int e = blockIdx.x * blockDim.x + threadIdx.x;
    if (e < E) atomicAdd(&deg[row[e]], 1u);
}

__global__ void dinv_kernel(const unsigned* __restrict__ deg, float* __restrict__ dinv, int N) {
    int i = blockIdx.x * blockDim.x + threadIdx.x;
    if (i < N) {
        unsigned d = deg[i];
        dinv[i] = d ? rsqrtf((float)d) : 0.0f;
    }
}

// ---------------- pack W1_0|W1_1 into WMMA B-operand lane order ----------------
// Bpk[sel][kk][lane][0..1]: lane = kh*16+m holds W[4*kk+2*kh + {0,1}][m].
// Inner gemm loop then does ONE coalesced b64 load per step for B.
__global__ void pack_w(const float* __restrict__ W0, const float* __restrict__ W1,
                       float* __restrict__ Bpk) {
    int t = blockIdx.x * blockDim.x + threadIdx.x;     // t < 2*KSTEP*32
    if (t >= 2 * KSTEP * 32) return;
    int lane = t & 31;
    int kk   = (t >> 5) % KSTEP;
    int sel  = t / (KSTEP * 32);
    const float* __restrict__ Wp = sel ? W1 : W0;
    int m  = lane & 15;
    int kh = lane >> 4;
    int krow = 4 * kk + 2 * kh;
    Bpk[(size_t)t * 2 + 0] = Wp[(size_t)krow * HID + m];
    Bpk[(size_t)t * 2 + 1] = Wp[(size_t)(krow + 1) * HID + m];
}

// ---------------- fused GEMM1: xw[:,0:16]=x@W1_0, xw[:,16:32]=x@W1_1 ----------------
// One wave per 16x16 output tile, V_WMMA_F32_16X16X4_F32 over K=500 (125 steps).
__global__ void wmma_gemm1(const float* __restrict__ x,
                           const float* __restrict__ Bpk,
                           float* __restrict__ xw, int N, int nTiles) {
    int wave = (int)((blockIdx.x * blockDim.x + threadIdx.x) >> 5);
    if (wave >= nTiles) return;               // wave-uniform: EXEC stays all-ones
    int lane    = threadIdx.x & 31;
    int rowTile = wave >> 1;
    int colSel  = wave & 1;
    int node0 = rowTile * 16;
    int m  = lane & 15;                       // M index (and N index for B/C)
    int kh = lane >> 4;                       // K half: 0 -> K=0,1 ; 1 -> K=2,3

    int arowIdx = node0 + m; if (arowIdx >= N) arowIdx = N - 1;   // clamp (stores guarded)
    const float* __restrict__ arow = x + (size_t)arowIdx * F_IN + 2 * kh;
    const float* __restrict__ brow = Bpk + (size_t)colSel * (KSTEP * 64) + lane * 2;

    v8f acc = {0.f, 0.f, 0.f, 0.f, 0.f, 0.f, 0.f, 0.f};
    for (int kk = 0; kk < KSTEP; ++kk) {
        v2f a = *(const v2f*)(arow + 4 * kk);          // x[row][4kk+2kh .. +1]
        v2f b = *(const v2f*)(brow + (size_t)kk * 64); // packed, fully coalesced
        acc = __builtin_amdgcn_wmma_f32_16x16x4_f32(
            /*neg_a=*/false, a, /*neg_b=*/false, b,
            /*c_mod=*/(short)0, acc, /*reuse_a=*/false, /*reuse_b=*/false);
    }
    // D layout: VGPR g -> M = g + 8*kh, N = m
    float* __restrict__ orow = xw + (size_t)(node0 + 8 * kh) * 32 + colSel * 16 + m;
    if (node0 + 16 <= N) {                    // uniform full-tile fast path
#pragma unroll
        for (int g = 0; g < 8; ++g) orow[(size_t)g * 32] = acc[g];
    } else {
#pragma unroll
        for (int g = 0; g < 8; ++g) {
            if (node0 + 8 * kh + g < N) orow[(size_t)g * 32] = acc[g];
        }
    }
}

// ---------------- edge scatter, layer 1 (16 features / edge) ----------------
__global__ void scatter1(const int* __restrict__ row, const int* __restrict__ col,
                         const float* __restrict__ dinv, const float* __restrict__ xw,
                         float* __restrict__ s1, long E) {
    long tid = (long)blockIdx.x * blockDim.x + threadIdx.x;
    long e = tid >> 4;
    int  j = (int)(tid & 15);
    if (e >= E) return;
    int r = row[e], c = col[e];
    float w = -dinv[r] * dinv[c];
    atomicAdd(&s1[(size_t)r * 16 + j], w * xw[(size_t)c * 32 + 16 + j]);
}

// ---------------- per-node: h = relu(...), g0 = h@W2_0, hw = h@W2_1 ----------------
__global__ void node_mid(const float* __restrict__ xw, const float* __restrict__ s1,
                         const float* __restrict__ b1,
                         const float* __restrict__ W20, const float* __restrict__ W21,
                         float* __restrict__ g0, float* __restrict__ hw, int N) {
    int i = blockIdx.x * blockDim.x + threadIdx.x;
    if (i >= N) return;
    float h[HID];
#pragma unroll
    for (int j = 0; j < HID; ++j) {
        float v = xw[(size_t)i * 32 + j] + s1[(size_t)i * 16 + j] + b1[j];
        h[j] = v > 0.f ? v : 0.f;
    }
#pragma unroll
    for (int o = 0; o < NCLS; ++o) {
        float a = 0.f, b = 0.f;
#pragma unroll
        for (int j = 0; j < HID; ++j) {
            a += h[j] * W20[j * NCLS + o];
            b += h[j] * W21[j * NCLS + o];
        }
        g0[(size_t)i * 8 + o] = a;
        hw[(size_t)i * 8 + o] = b;
    }
}

// ---------------- edge scatter, layer 2 (7 features / edge, stride-8 pad) ----------------
__global__ void scatter2(const int* __restrict__ row, const int* __restrict__ col,
                         const float* __restrict__ dinv, const float* __restrict__ hw,
                         float* __restrict__ s2, long E) {
    long tid = (long)blockIdx.x * blockDim.x + threadIdx.x;
    long e = tid >> 3;
    int  j = (int)(tid & 7);
    if (e >= E || j >= NCLS) return;
    int r = row[e], c = col[e];
    float w = -dinv[r] * dinv[c];
    atomicAdd(&s2[(size_t)r * 8 + j], w * hw[(size_t)c * 8 + j]);
}

// ---------------- out = log_softmax(g0 + s2 + b2) ----------------
__global__ void out_kernel(const float* __restrict__ g0, const float* __restrict__ s2,
                           const float* __restrict__ b2, float* __restrict__ out, int N) {
    int i = blockIdx.x * blockDim.x + threadIdx.x;
    if (i >= N) return;
    float o[NCLS];
#pragma unroll
    for (int j = 0; j < NCLS; ++j) o[j] = g0[(size_t)i * 8 + j] + s2[(size_t)i * 8 + j] + b2[j];
    float m = o[0];
#pragma unroll
    for (int j = 1; j < NCLS; ++j) m = o[j] > m ? o[j] : m;
    float s = 0.f;
#pragma unroll
    for (int j = 0; j < NCLS; ++j) s += expf(o[j] - m);
    float l = logf(s);
#pragma unroll
    for (int j = 0; j < NCLS; ++j) out[(size_t)i * NCLS + j] = o[j] - m - l;
}

extern "C" void kernel_launch(void* const* d_in, const int* in_sizes, int n_in,
                              void* d_out, int out_size, void* d_ws, size_t ws_size,
                              hipStream_t stream) {
    const float* x    = (const float*)d_in[0];
    const int*   eidx = (const int*)d_in[1];
    const float* W1_0 = (const float*)d_in[2];
    const float* W1_1 = (const float*)d_in[3];
    const float* b1   = (const float*)d_in[4];
    const float* W2_0 = (const float*)d_in[5];
    const float* W2_1 = (const float*)d_in[6];
    const float* b2   = (const float*)d_in[7];
    float* out = (float*)d_out;

    const int N = in_sizes[0] / F_IN;
    const long E = in_sizes[1] / 2;
    const int* row = eidx;
    const int* col = eidx + E;

    // workspace carve-up (256B aligned)
    char* ws = (char*)d_ws;
    size_t off = 0;
    auto carve = [&](size_t bytes) { char* p = ws + off; off = (off + bytes + 255) & ~(size_t)255; return p; };
    unsigned* deg  = (unsigned*)carve((size_t)N * 4);
    float*    dinv = (float*)   carve((size_t)N * 4);
    float*    xw   = (float*)   carve((size_t)N * 32 * 4);
    float*    s1   = (float*)   carve((size_t)N * 16 * 4);
    float*    g0   = (float*)   carve((size_t)N * 8 * 4);
    float*    hw   = (float*)   carve((size_t)N * 8 * 4);
    float*    s2   = (float*)   carve((size_t)N * 8 * 4);
    float*    Bpk  = (float*)   carve((size_t)2 * KSTEP * 64 * 4);   // 64 KB packed B
    (void)ws_size;

    // zero accumulators (graph-capturable async memsets)
    hipMemsetAsync(deg, 0, (size_t)N * 4, stream);
    hipMemsetAsync(s1,  0, (size_t)N * 16 * 4, stream);
    hipMemsetAsync(s2,  0, (size_t)N * 8 * 4, stream);

    // degree + dinv
    deg_kernel<<<(int)((E + 255) / 256), 256, 0, stream>>>(row, deg, (int)E);
    dinv_kernel<<<(N + 255) / 256, 256, 0, stream>>>(deg, dinv, N);

    // pack weights into WMMA B lane order (8000 threads, one-shot)
    pack_w<<<(2 * KSTEP * 32 + 255) / 256, 256, 0, stream>>>(W1_0, W1_1, Bpk);

    // fused GEMM1 via WMMA f32: 16x16 tiles, 2 col-tiles per 16-row group
    int nRowTiles = (N + 15) / 16;
    int nTiles = nRowTiles * 2;
    int gemmBlocks = (nTiles + 3) / 4;   // 4 waves (128 threads) per block
    wmma_gemm1<<<gemmBlocks, 128, 0, stream>>>(x, Bpk, xw, N, nTiles);

    // scatter layer 1: one thread per (edge, feature)
    long t1 = E * 16;
    scatter1<<<(int)((t1 + 255) / 256), 256, 0, stream>>>(row, col, dinv, xw, s1, E);

    // per-node relu + both 16x7 projections
    node_mid<<<(N + 255) / 256, 256, 0, stream>>>(xw, s1, b1, W2_0, W2_1, g0, hw, N);

    // scatter layer 2: one thread per (edge, padded-feature)
    long t2 = E * 8;
    scatter2<<<(int)((t2 + 255) / 256), 256, 0, stream>>>(row, col, dinv, hw, s2, E);

    // final bias + log_softmax
    out_kernel<<<(N + 255) / 256, 256, 0, stream>>>(g0, s2, b2, out, N);
}